// RGATCellStack_59210419143207
// MI455X (gfx1250) — compile-verified
//
#include <hip/hip_runtime.h>

// CDNA5 / gfx1250, wave32.
typedef __attribute__((ext_vector_type(2))) float v2f;
typedef __attribute__((ext_vector_type(8))) float v8f;

#define DFEAT 64
#define NEG_SLOPE 0.2f
#define EPSV 1e-9f

__device__ __forceinline__ void atomicMaxFloat(float* addr, float val) {
  // Monotone float max via int/uint atomics (valid with -inf init).
  if (val >= 0.0f) {
    atomicMax((int*)addr, __float_as_int(val));
  } else {
    atomicMin((unsigned int*)addr, __float_as_uint(val));
  }
}

// ---------------------------------------------------------------- init ------
__global__ void rgat_init(float* __restrict__ m, float* __restrict__ denom,
                          float* __restrict__ red, int N) {
  int idx = blockIdx.x * blockDim.x + threadIdx.x;
  if (idx < N * DFEAT) red[idx] = 0.0f;
  if (idx < N) {
    m[idx] = -__builtin_inff();
    denom[idx] = 0.0f;
  }
}

// ------------------------------------------------- hW = h @ W_r (WMMA) ------
// One wave computes a 16x64 strip of hW for one relation r.
// V_WMMA_F32_16X16X4_F32 layouts (ISA 7.12.2):
//   A 16x4 : lane L holds row M=L&15, K = (L>=16 ? 2 : 0) + vgpr
//   B 4x16 : lane L holds col N=L&15, K = (L>=16 ? 2 : 0) + vgpr
//   C 16x16: lane L holds col N=L&15, rows M = (L>=16 ? 8 : 0) + vgpr(0..7)
__global__ void rgat_hw(const float* __restrict__ h, const float* __restrict__ W,
                        float* __restrict__ hW, int N, int R, int ntiles) {
  int wave = blockIdx.x * (blockDim.x >> 5) + (threadIdx.x >> 5);
  int lane = threadIdx.x & 31;
  if (wave >= R * ntiles) return;
  int r = wave / ntiles;
  int tile = wave - r * ntiles;
  int n0 = tile * 16;

  const float* Wr = W + (size_t)r * DFEAT * DFEAT;

  int half = lane >> 4;           // 0: lanes 0-15, 1: lanes 16-31
  int lid16 = lane & 15;
  int kOfs = half * 2;            // K offset for this lane half (A and B)

  int rowA = n0 + lid16;
  if (rowA >= N) rowA = N - 1;    // clamp (store is guarded)
  const float* hrow = h + (size_t)rowA * DFEAT;

  v8f acc0 = {}, acc1 = {}, acc2 = {}, acc3 = {};

#pragma unroll
  for (int kb = 0; kb < DFEAT; kb += 4) {
    v2f a;
    a.x = hrow[kb + kOfs + 0];
    a.y = hrow[kb + kOfs + 1];
    const float* Wk0 = Wr + (size_t)(kb + kOfs) * DFEAT;  // row k
    const float* Wk1 = Wk0 + DFEAT;                       // row k+1
    v2f b0; b0.x = Wk0[ 0 + lid16]; b0.y = Wk1[ 0 + lid16];
    v2f b1; b1.x = Wk0[16 + lid16]; b1.y = Wk1[16 + lid16];
    v2f b2; b2.x = Wk0[32 + lid16]; b2.y = Wk1[32 + lid16];
    v2f b3; b3.x = Wk0[48 + lid16]; b3.y = Wk1[48 + lid16];
    acc0 = __builtin_amdgcn_wmma_f32_16x16x4_f32(false, a, false, b0, (short)0, acc0, false, false);
    acc1 = __builtin_amdgcn_wmma_f32_16x16x4_f32(false, a, false, b1, (short)0, acc1, false, false);
    acc2 = __builtin_amdgcn_wmma_f32_16x16x4_f32(false, a, false, b2, (short)0, acc2, false, false);
    acc3 = __builtin_amdgcn_wmma_f32_16x16x4_f32(false, a, false, b3, (short)0, acc3, false, false);
  }

  int mBase = half * 8;
  float* outR = hW + (size_t)r * N * DFEAT;
#pragma unroll
  for (int j = 0; j < 8; ++j) {
    int row = n0 + mBase + j;
    if (row < N) {
      float* orow = outR + (size_t)row * DFEAT + lid16;
      orow[ 0] = acc0[j];
      orow[16] = acc1[j];
      orow[32] = acc2[j];
      orow[48] = acc3[j];
    }
  }
}

// -------------------------------------------- edge logits + segment max -----
__global__ void rgat_logit(const float* __restrict__ h, const float* __restrict__ hW,
                           const float* __restrict__ a,
                           const int* __restrict__ src, const int* __restrict__ dst,
                           const int* __restrict__ etype,
                           float* __restrict__ e_out, float* __restrict__ m_out,
                           int N, int E) {
  int wave = blockIdx.x * (blockDim.x >> 5) + (threadIdx.x >> 5);
  int lane = threadIdx.x & 31;
  if (wave >= E) return;
  int s = src[wave], d = dst[wave], r = etype[wave];
  const float* hd  = h + (size_t)d * DFEAT;
  const float* msg = hW + ((size_t)r * N + (size_t)s) * DFEAT;
  const float* ar  = a + (size_t)r * 2 * DFEAT;

  float sum = hd[lane] * ar[lane] + hd[lane + 32] * ar[lane + 32]
            + msg[lane] * ar[DFEAT + lane] + msg[lane + 32] * ar[DFEAT + lane + 32];
#pragma unroll
  for (int off = 16; off > 0; off >>= 1) sum += __shfl_down(sum, off, 32);

  if (lane == 0) {
    float e = sum > 0.0f ? sum : NEG_SLOPE * sum;
    e_out[wave] = e;
    atomicMaxFloat(m_out + d, e);
  }
}

// ------------------------------------- exp weights + segment sums -----------
__global__ void rgat_accum(const float* __restrict__ hW, const float* __restrict__ e_in,
                           const float* __restrict__ m_in,
                           const int* __restrict__ src, const int* __restrict__ dst,
                           const int* __restrict__ etype,
                           float* __restrict__ denom, float* __restrict__ red,
                           int N, int E) {
  int wave = blockIdx.x * (blockDim.x >> 5) + (threadIdx.x >> 5);
  int lane = threadIdx.x & 31;
  if (wave >= E) return;
  int s = src[wave], d = dst[wave], r = etype[wave];
  float w = __expf(e_in[wave] - m_in[d]);
  const float* msg = hW + ((size_t)r * N + (size_t)s) * DFEAT;
  float* rd = red + (size_t)d * DFEAT;
  atomicAdd(rd + lane,      w * msg[lane]);
  atomicAdd(rd + lane + 32, w * msg[lane + 32]);
  if (lane == 0) atomicAdd(denom + d, w);
}

// --------------------------------------------------------- finalize ---------
__global__ void rgat_final(const float* __restrict__ h, const float* __restrict__ mask,
                           const float* __restrict__ red, const float* __restrict__ denom,
                           float* __restrict__ out, int N) {
  int idx = blockIdx.x * blockDim.x + threadIdx.x;
  if (idx >= N * DFEAT) return;
  int n = idx >> 6;       // / DFEAT
  int dd = idx & (DFEAT - 1);
  out[idx] = h[idx] + red[idx] / (denom[n] + EPSV) * mask[dd];
}

extern "C" void kernel_launch(void* const* d_in, const int* in_sizes, int n_in,
                              void* d_out, int out_size, void* d_ws, size_t ws_size,
                              hipStream_t stream) {
  (void)n_in; (void)out_size; (void)ws_size;
  const float* h     = (const float*)d_in[0];
  const float* W     = (const float*)d_in[1];
  const float* a     = (const float*)d_in[2];
  const float* mask  = (const float*)d_in[3];
  const int*   src   = (const int*)d_in[4];
  const int*   dst   = (const int*)d_in[5];
  const int*   etype = (const int*)d_in[6];
  float* out = (float*)d_out;

  const int N = in_sizes[0] / DFEAT;
  const int R = in_sizes[1] / (DFEAT * DFEAT);
  const int E = in_sizes[4];

  // workspace layout (floats)
  float* ws = (float*)d_ws;
  size_t off = 0;
  float* hW    = ws + off; off += (size_t)R * N * DFEAT;
  float* e_buf = ws + off; off += (size_t)E;
  float* m_buf = ws + off; off += (size_t)N;
  float* denom = ws + off; off += (size_t)N;
  float* red   = ws + off; off += (size_t)N * DFEAT;

  const int TPB = 256;
  const int wavesPerBlock = TPB / 32;

  // 1. init
  int nd = N * DFEAT;
  rgat_init<<<(nd + TPB - 1) / TPB, TPB, 0, stream>>>(m_buf, denom, red, N);

  // 2. hW = h @ W_r (WMMA)
  int ntiles = (N + 15) / 16;
  int hwWaves = R * ntiles;
  rgat_hw<<<(hwWaves + wavesPerBlock - 1) / wavesPerBlock, TPB, 0, stream>>>(
      h, W, hW, N, R, ntiles);

  // 3. logits + segment max
  rgat_logit<<<(E + wavesPerBlock - 1) / wavesPerBlock, TPB, 0, stream>>>(
      h, hW, a, src, dst, etype, e_buf, m_buf, N, E);

  // 4. exp + segment sums
  rgat_accum<<<(E + wavesPerBlock - 1) / wavesPerBlock, TPB, 0, stream>>>(
      hW, e_buf, m_buf, src, dst, etype, denom, red, N, E);

  // 5. finalize
  rgat_final<<<(nd + TPB - 1) / TPB, TPB, 0, stream>>>(h, mask, red, denom, out, N);
}